// Model_21689584844831
// MI455X (gfx1250) — compile-verified
//
#include <hip/hip_runtime.h>

typedef float v2f __attribute__((ext_vector_type(2)));
typedef float v8f __attribute__((ext_vector_type(8)));

// ---------------- utility kernels ----------------

__global__ void zero_f32(float* __restrict__ p, long long n) {
  long long i = (long long)blockIdx.x * blockDim.x + threadIdx.x;
  long long stride = (long long)gridDim.x * blockDim.x;
  for (; i < n; i += stride) p[i] = 0.0f;
}

__global__ void mean_reduce(const float* __restrict__ x, int N,
                            float* __restrict__ sums) {
  __shared__ float sm[3][256];
  float a0 = 0.f, a1 = 0.f, a2 = 0.f;
  int i = blockIdx.x * blockDim.x + threadIdx.x;
  int stride = gridDim.x * blockDim.x;
  for (; i < N; i += stride) {
    a0 += x[3 * i + 0];
    a1 += x[3 * i + 1];
    a2 += x[3 * i + 2];
  }
  int t = threadIdx.x;
  sm[0][t] = a0; sm[1][t] = a1; sm[2][t] = a2;
  __syncthreads();
  for (int s = 128; s > 0; s >>= 1) {
    if (t < s) {
      sm[0][t] += sm[0][t + s];
      sm[1][t] += sm[1][t + s];
      sm[2][t] += sm[2][t + s];
    }
    __syncthreads();
  }
  if (t == 0) {
    atomicAdd(&sums[0], sm[0][0]);
    atomicAdd(&sums[1], sm[1][0]);
    atomicAdd(&sums[2], sm[2][0]);
  }
}

// x0[N,4] = x[N,3] - mean, padded 4th column = 0
__global__ void center_kernel(const float* __restrict__ x,
                              const float* __restrict__ sums, float invN,
                              float* __restrict__ x0, int N) {
  int i = blockIdx.x * blockDim.x + threadIdx.x;
  if (i >= N) return;
  x0[4 * i + 0] = x[3 * i + 0] - sums[0] * invN;
  x0[4 * i + 1] = x[3 * i + 1] - sums[1] * invN;
  x0[4 * i + 2] = x[3 * i + 2] - sums[2] * invN;
  x0[4 * i + 3] = 0.0f;
}

// ---------------- edge scatter (segment-sum) ----------------
// Gathers the *virtual* concatenation [A | B | C] at row src and
// atomically accumulates into agg[dst]. WPE threads cooperate per edge;
// consecutive lanes share src/dst cachelines, and both the gather reads
// and the f32 atomic adds stay resident in the 192MB L2.
template <int WPE>
__global__ void scatter_kernel(const int* __restrict__ ei, int E,
                               const float* __restrict__ A, int nA, int sA,
                               const float* __restrict__ B, int nB, int sB,
                               const float* __restrict__ C, int nC, int sC,
                               float* __restrict__ agg, int sAgg) {
  long long t = (long long)blockIdx.x * blockDim.x + threadIdx.x;
  long long e = t / WPE;
  int r = (int)(t % WPE);
  if (e >= E) return;
  int src = ei[e];
  int dst = ei[(long long)E + e];
  int nTot = nA + nB + nC;
  float* dstRow = agg + (long long)dst * sAgg;
  for (int f = r; f < nTot; f += WPE) {
    float v;
    if (f < nA)            v = A[(long long)src * sA + f];
    else if (f < nA + nB)  v = B[(long long)src * sB + (f - nA)];
    else                   v = C[(long long)src * sC + (f - nA - nB)];
    atomicAdd(&dstRow[f], v);
  }
}

// ---------------- WMMA f32 GEMM: out = in[N,KPAD] @ W[outRows,Kraw]^T + bias ----------------
// One block = CTILES waves; each wave owns one 16-wide output-column tile of
// a 16-node row slab. A-tile and zero-padded W are staged in LDS; the K loop
// issues V_WMMA_F32_16X16X4_F32 per 4-deep K step.
// A layout: lane m=lane&15 holds row m; VGPR0/1 = K pair, lane half selects K+0/+2.
// B layout: lane holds column n=lane&15 of tile; VGPR0/1 = K rows (khalf*2, +1).
// D layout: VGPR v -> (row v + 8*khalf, col lane&15).
template <int KPAD, int CTILES>
__global__ __launch_bounds__(CTILES * 32)
void gemm_wmma(const float* __restrict__ in, const float* __restrict__ W,
               int Kraw, int outRows, const float* __restrict__ bias,
               float* __restrict__ out, int outStride, int outCols,
               const float* __restrict__ meanSums, float invN, int N) {
  __shared__ float Wl[CTILES * 16 * KPAD];
  __shared__ float Al[16 * KPAD];
  const int tid = threadIdx.x;
  const int nthr = CTILES * 32;

  for (int idx = tid; idx < CTILES * 16 * KPAD; idx += nthr) {
    int n = idx / KPAD;
    int k = idx % KPAD;
    Wl[idx] = (n < outRows && k < Kraw) ? W[n * Kraw + k] : 0.0f;
  }
  const int row0 = blockIdx.x * 16;
  for (int idx = tid; idx < 16 * KPAD; idx += nthr) {
    int rr = idx / KPAD;
    Al[idx] = (row0 + rr < N) ? in[(long long)(row0 + rr) * KPAD + (idx % KPAD)]
                              : 0.0f;
  }
  __syncthreads();

  const int wave = tid >> 5;
  const int lane = tid & 31;
  const int m = lane & 15;
  const int khalf = lane >> 4;
  const float* Wt = &Wl[wave * 16 * KPAD];

  v8f c = {};
#pragma unroll
  for (int k0 = 0; k0 < KPAD; k0 += 4) {
    v2f a, b;
    a[0] = Al[m * KPAD + k0 + khalf * 2 + 0];
    a[1] = Al[m * KPAD + k0 + khalf * 2 + 1];
    b[0] = Wt[m * KPAD + k0 + khalf * 2 + 0];
    b[1] = Wt[m * KPAD + k0 + khalf * 2 + 1];
    c = __builtin_amdgcn_wmma_f32_16x16x4_f32(
        /*neg_a=*/false, a, /*neg_b=*/false, b,
        /*c_mod=*/(short)0, c, /*reuse_a=*/false, /*reuse_b=*/false);
  }

  const int ncol = wave * 16 + m;
  if (ncol < outCols) {
    float bv = bias[ncol];
    if (meanSums) bv += meanSums[ncol] * invN;
#pragma unroll
    for (int v = 0; v < 8; ++v) {
      int rr = v + khalf * 8;
      if (row0 + rr < N)
        out[(long long)(row0 + rr) * outStride + ncol] = c[v] + bv;
    }
  }
}

// fixed-node overwrite: out[idx] = x[idx] for the two static index sets
__global__ void fixed_kernel(const float* __restrict__ x,
                             float* __restrict__ out) {
  int i = blockIdx.x * blockDim.x + threadIdx.x;
  const int HALF = 49 * 14;
  if (i >= 2 * HALF) return;
  int j = i % HALF;
  int g = j / 14;
  int r = j % 14;
  int idx = (i < HALF) ? (g * 40 + r) : (g * 40 + 25 + r);
  out[3 * idx + 0] = x[3 * idx + 0];
  out[3 * idx + 1] = x[3 * idx + 1];
  out[3 * idx + 2] = x[3 * idx + 2];
}

// ---------------- launcher ----------------

extern "C" void kernel_launch(void* const* d_in, const int* in_sizes, int n_in,
                              void* d_out, int out_size, void* d_ws,
                              size_t ws_size, hipStream_t stream) {
  const float* x     = (const float*)d_in[0];
  const int*   ei    = (const int*)d_in[1];
  const float* c1_w2 = (const float*)d_in[5];
  const float* c1_b2 = (const float*)d_in[6];
  const float* c2_w2 = (const float*)d_in[9];
  const float* c2_b2 = (const float*)d_in[10];
  const float* c3_w2 = (const float*)d_in[13];
  const float* c3_b2 = (const float*)d_in[14];
  float* out = (float*)d_out;

  const int N = in_sizes[0] / 3;
  const int E = in_sizes[1] / 2;
  const float invN = 1.0f / (float)N;

  // workspace layout (floats): [mean:16][x0:N*4][g1:N*52][g2:N*52][agg:N*104]
  float* ws   = (float*)d_ws;
  float* mean = ws;
  float* x0   = ws + 16;
  float* g1   = x0 + (size_t)N * 4;
  float* g2   = g1 + (size_t)N * 52;
  float* agg  = g2 + (size_t)N * 52;

  const int TB = 256;
  const int rowBlocks = (N + 15) / 16;

  // mean + centering
  zero_f32<<<1, 64, 0, stream>>>(mean, 16);
  mean_reduce<<<512, TB, 0, stream>>>(x, N, mean);
  center_kernel<<<(N + TB - 1) / TB, TB, 0, stream>>>(x, mean, invN, x0, N);

  // ---- layer 1: agg[N,4] = scatter(x0); g1 = agg @ c1_w2^T + b ----
  zero_f32<<<2048, TB, 0, stream>>>(agg, (long long)N * 4);
  {
    unsigned long long T = (unsigned long long)E * 4ull;
    scatter_kernel<4><<<(unsigned)((T + TB - 1) / TB), TB, 0, stream>>>(
        ei, E, x0, 3, 4, nullptr, 0, 0, nullptr, 0, 0, agg, 4);
  }
  gemm_wmma<4, 4><<<rowBlocks, 4 * 32, 0, stream>>>(
      agg, c1_w2, 3, 50, c1_b2, g1, 52, 50, nullptr, invN, N);

  // ---- layer 2: agg[N,56] = scatter([g1|x0]); g2 = agg @ c2_w2^T + b ----
  zero_f32<<<2048, TB, 0, stream>>>(agg, (long long)N * 56);
  {
    unsigned long long T = (unsigned long long)E * 8ull;
    scatter_kernel<8><<<(unsigned)((T + TB - 1) / TB), TB, 0, stream>>>(
        ei, E, g1, 50, 52, x0, 3, 4, nullptr, 0, 0, agg, 56);
  }
  gemm_wmma<56, 4><<<rowBlocks, 4 * 32, 0, stream>>>(
      agg, c2_w2, 53, 50, c2_b2, g2, 52, 50, nullptr, invN, N);

  // ---- layer 3: agg[N,104] = scatter([g2|g1|x0]); out = agg @ c3_w2^T + b + mean ----
  zero_f32<<<2048, TB, 0, stream>>>(agg, (long long)N * 104);
  {
    unsigned long long T = (unsigned long long)E * 16ull;
    scatter_kernel<16><<<(unsigned)((T + TB - 1) / TB), TB, 0, stream>>>(
        ei, E, g2, 50, 52, g1, 50, 52, x0, 3, 4, agg, 104);
  }
  gemm_wmma<104, 1><<<rowBlocks, 32, 0, stream>>>(
      agg, c3_w2, 103, 3, c3_b2, out, 3, 3, mean, invN, N);

  // fixed-node overwrite
  fixed_kernel<<<(2 * 49 * 14 + TB - 1) / TB, TB, 0, stream>>>(x, out);
}